// ETGNN_87917980549282
// MI455X (gfx1250) — compile-verified
//
#include <hip/hip_runtime.h>
#include <hip/hip_bf16.h>
#include <cmath>

typedef __attribute__((ext_vector_type(2))) float v2f;
typedef __attribute__((ext_vector_type(8))) float v8f;

#define BM 128
#define BN 64
#define KT 32
#define TPB 256
#define LDA (KT + 2)        // even stride: aligned b64 A-fragment loads, bank-spread
#define LDB2 (2 * BN + 32)  // paired-k B rows; +32 floats => kh halves hit disjoint banks

// ---------------------------------------------------------------------------
// Virtual A-operand loader.
// MODE 0: sin time features  A[e][k] = sin(ts[e] * s^(-k/s))
// MODE 1: plain              A[r][k] = Aplain[r*D + k]           (K == D)
// MODE 2: node-fc input      A[n][k] = k<3D ? agg[n][k]*invdeg[n] : self[n][k-3D]
// MODE 3: edge-fc input      A[e][k] = [ hn[src], efeat, ts_enc, hn[dst] ]
// RELU_E: relu applied to self-feature (MODE 2) / edge-feature (MODE 3) on read.
// ---------------------------------------------------------------------------
template <int MODE, bool RELU_E>
__device__ __forceinline__ float load_a(
    int row, int k, int D,
    const float* __restrict__ Aplain,
    const float* __restrict__ agg, const float* __restrict__ selfF,
    const float* __restrict__ invdeg,
    const float* __restrict__ hn, const float* __restrict__ ef,
    const float* __restrict__ te,
    const int* __restrict__ srcIdx, const int* __restrict__ dstIdx,
    const float* __restrict__ ts, float lg)
{
    if (MODE == 0) {
        float f = expf(-(float)k * lg);               // s^(-k/s)
        return sinf(ts[row] * f);
    } else if (MODE == 1) {
        return Aplain[(size_t)row * D + k];
    } else if (MODE == 2) {
        if (k < 3 * D) return agg[(size_t)row * 3 * D + k] * invdeg[row];
        float v = selfF[(size_t)row * D + (k - 3 * D)];
        if (RELU_E) v = fmaxf(v, 0.0f);
        return v;
    } else {
        if (k < D)      return hn[(size_t)srcIdx[row] * D + k];
        if (k < 2 * D) { float v = ef[(size_t)row * D + (k - D)];
                         if (RELU_E) v = fmaxf(v, 0.0f); return v; }
        if (k < 3 * D)  return te[(size_t)row * D + (k - 2 * D)];
        return hn[(size_t)dstIdx[row] * D + (k - 3 * D)];
    }
}

// ---------------------------------------------------------------------------
// Tiled fp32 GEMM:  C[M x Nn] = act( A[M x K] @ W[K x Nn] + bias )
// 256 threads = 8 wave32; each wave owns a 16x64 strip (4 accumulators) of the
// 128x64 block tile, built from V_WMMA_F32_16X16X4_F32. LDS double-buffered.
// B is stored k-paired in LDS so each fragment is one aligned ds_load_b64.
// ---------------------------------------------------------------------------
template <int MODE, bool RELU_OUT, bool RELU_E>
__global__ __launch_bounds__(TPB) void gemm_kernel(
    const float* __restrict__ Aplain,
    const float* __restrict__ agg, const float* __restrict__ selfF,
    const float* __restrict__ invdeg,
    const float* __restrict__ hn, const float* __restrict__ ef,
    const float* __restrict__ te,
    const int* __restrict__ srcIdx, const int* __restrict__ dstIdx,
    const float* __restrict__ ts,
    const float* __restrict__ W, const float* __restrict__ bias,
    float* __restrict__ C,
    int M, int K, int Nn, int D, float lg)
{
    __shared__ float sA[2][BM][LDA];
    __shared__ float sBp[2][KT / 2][LDB2];

    const int tid  = threadIdx.x;
    const int wave = tid >> 5;
    const int lane = tid & 31;
    const int lm   = lane & 15;   // M (for A) / N (for B) sub-index
    const int kh   = lane >> 4;   // K half-select per ISA fragment layout
    const int m0   = blockIdx.x * BM;
    const int n0   = blockIdx.y * BN;
    const int mrow = wave * 16;

    auto stage = [&](int buf, int k0) {
        // --- A tile: k-contiguous per row -> coalesced virtual gathers ---
        for (int i = tid; i < BM * KT; i += TPB) {
            int m = i / KT, kk = i % KT;
            int gm = m0 + m, gk = k0 + kk;
            float v = 0.0f;
            if (gm < M && gk < K)
                v = load_a<MODE, RELU_E>(gm, gk, D, Aplain, agg, selfF, invdeg,
                                         hn, ef, te, srcIdx, dstIdx, ts, lg);
            sA[buf][m][kk] = v;
        }
        // --- B tile: store k-pairs adjacently => fragment = 1x ds_load_b64 ---
        for (int i = tid; i < (KT / 2) * BN; i += TPB) {
            int kp = i / BN, n = i % BN;
            int gk = k0 + 2 * kp, gn = n0 + n;
            v2f v;
            v.x = (gk     < K && gn < Nn) ? W[(size_t)gk       * Nn + gn] : 0.0f;
            v.y = (gk + 1 < K && gn < Nn) ? W[(size_t)(gk + 1) * Nn + gn] : 0.0f;
            *(v2f*)&sBp[buf][kp][2 * n] = v;
            if (gk + 2 * KT < K && gn < Nn)   // keep next-next weight tile warm
                __builtin_prefetch(&W[(size_t)(gk + 2 * KT) * Nn + gn], 0, 1);
        }
    };

    v8f acc[4];
#pragma unroll
    for (int t = 0; t < 4; ++t) acc[t] = (v8f)0.0f;

    const int nStages = (K + KT - 1) / KT;
    stage(0, 0);
    __syncthreads();

    for (int st = 0; st < nStages; ++st) {
        const int cur = st & 1;
        if (st + 1 < nStages) stage(cur ^ 1, (st + 1) * KT);  // overlap w/ compute

#pragma unroll
        for (int kk = 0; kk < KT; kk += 4) {
            v2f a = *(const v2f*)&sA[cur][mrow + lm][kk + 2 * kh];
#pragma unroll
            for (int t = 0; t < 4; ++t) {
                v2f b = *(const v2f*)&sBp[cur][(kk >> 1) + kh][2 * (t * 16 + lm)];
                acc[t] = __builtin_amdgcn_wmma_f32_16x16x4_f32(
                    false, a, false, b, (short)0, acc[t], false, false);
            }
        }
        __syncthreads();   // next buffer written & current fully consumed
    }

    // --- epilogue: bias (+relu) and store; C row stride == Nn (packed) ---
#pragma unroll
    for (int t = 0; t < 4; ++t) {
        int col = n0 + t * 16 + lm;
#pragma unroll
        for (int r = 0; r < 8; ++r) {
            int row = m0 + mrow + r + 8 * kh;
            if (row < M && col < Nn) {
                float v = acc[t][r] + bias[col];
                if (RELU_OUT) v = fmaxf(v, 0.0f);
                C[(size_t)row * Nn + col] = v;
            }
        }
    }
}

// ---------------------------------------------------------------------------
// Scatter: agg[dst[e]] += [ h[src[e]] | efeat[e] | ts_enc[e] ]  (3D columns)
// ---------------------------------------------------------------------------
template <bool RELU>
__global__ void scatter_kernel(const float* __restrict__ hsrc,
                               const float* __restrict__ ef,
                               const float* __restrict__ te,
                               const int* __restrict__ src,
                               const int* __restrict__ dst,
                               float* __restrict__ agg, int E, int D)
{
    int e = blockIdx.x;
    if (e >= E) return;
    int s = src[e], d = dst[e];
    const float* hs = hsrc + (size_t)s * D;
    float* out = agg + (size_t)d * 3 * D;
    for (int k = threadIdx.x; k < 3 * D; k += blockDim.x) {
        float v;
        if (k < D)          { v = hs[k];                       if (RELU) v = fmaxf(v, 0.0f); }
        else if (k < 2 * D) { v = ef[(size_t)e * D + (k - D)]; if (RELU) v = fmaxf(v, 0.0f); }
        else                  v = te[(size_t)e * D + (k - 2 * D)];
        atomicAdd(&out[k], v);
    }
}

__global__ void fill_kernel(float* __restrict__ p, float v, size_t n) {
    size_t i = (size_t)blockIdx.x * blockDim.x + threadIdx.x;
    size_t stride = (size_t)gridDim.x * blockDim.x;
    for (; i < n; i += stride) p[i] = v;
}

__global__ void deg_kernel(const int* __restrict__ dst, float* __restrict__ deg, int E) {
    int e = blockIdx.x * blockDim.x + threadIdx.x;
    if (e < E) atomicAdd(&deg[dst[e]], 1.0f);
}

__global__ void invdeg_kernel(float* __restrict__ deg, int N) {
    int n = blockIdx.x * blockDim.x + threadIdx.x;
    if (n < N) deg[n] = 1.0f / fmaxf(deg[n], 1.0f);
}

extern "C" void kernel_launch(void* const* d_in, const int* in_sizes, int n_in,
                              void* d_out, int out_size, void* d_ws, size_t ws_size,
                              hipStream_t stream) {
    const int*   src   = (const int*)d_in[0];
    const int*   dst   = (const int*)d_in[1];
    const float* ts    = (const float*)d_in[2];
    const float* nfeat = (const float*)d_in[3];
    const float* efeat = (const float*)d_in[4];
    const float* te_w1 = (const float*)d_in[5];
    const float* te_b1 = (const float*)d_in[6];
    const float* te_w2 = (const float*)d_in[7];
    const float* te_b2 = (const float*)d_in[8];
    const float* wn1   = (const float*)d_in[9];
    const float* bn1   = (const float*)d_in[10];
    const float* we1   = (const float*)d_in[11];
    const float* be1   = (const float*)d_in[12];
    const float* wn2   = (const float*)d_in[13];
    const float* bn2   = (const float*)d_in[14];
    const float* we2   = (const float*)d_in[15];
    const float* be2   = (const float*)d_in[16];

    const int D = in_sizes[6];          // te_b1 length
    const int E = in_sizes[0];
    const int N = in_sizes[3] / D;
    const int n_train = out_size / D;
    const int s = (int)std::sqrt((double)D);
    const float lg = (float)(std::log((double)s) / (double)s);

    // workspace layout (floats)
    float* ws = (float*)d_ws;
    size_t o = 0;
    float* tsenc = ws + o; o += (size_t)E * D;
    float* e1    = ws + o; o += (size_t)E * D;   // doubles as TE-hidden temp
    float* h1    = ws + o; o += (size_t)N * D;
    float* h2    = ws + o; o += (size_t)N * D;
    float* agg   = ws + o; o += (size_t)N * 3 * D;
    float* deg   = ws + o; o += (size_t)N;
    (void)ws_size;

    float* out = (float*)d_out;

    const dim3 blk(TPB);
    const dim3 gE((E + BM - 1) / BM, (D + BN - 1) / BN);
    const dim3 gN((N + BM - 1) / BM, (D + BN - 1) / BN);
    const dim3 gT((n_train + BM - 1) / BM, (D + BN - 1) / BN);

    // degrees -> invdeg (shared by both layers)
    fill_kernel<<<256, 256, 0, stream>>>(deg, 0.0f, (size_t)N);
    deg_kernel<<<(E + 255) / 256, 256, 0, stream>>>(dst, deg, E);
    invdeg_kernel<<<(N + 255) / 256, 256, 0, stream>>>(deg, N);

    // time encoder: hidden = relu(sin(ts*w) @ te_w1 + b1); tsenc = hidden @ te_w2 + b2
    gemm_kernel<0, true, false><<<gE, blk, 0, stream>>>(
        nullptr, nullptr, nullptr, nullptr, nullptr, nullptr, nullptr,
        nullptr, nullptr, ts, te_w1, te_b1, e1, E, D, D, D, lg);
    gemm_kernel<1, false, false><<<gE, blk, 0, stream>>>(
        e1, nullptr, nullptr, nullptr, nullptr, nullptr, nullptr,
        nullptr, nullptr, nullptr, te_w2, te_b2, tsenc, E, D, D, D, lg);

    // ---- layer 1 ----
    fill_kernel<<<2048, 256, 0, stream>>>(agg, 0.0f, (size_t)N * 3 * D);
    scatter_kernel<false><<<E, 256, 0, stream>>>(nfeat, efeat, tsenc, src, dst, agg, E, D);
    // hn1 (raw, pre-relu) = [mean | nfeat] @ wn1 + bn1
    gemm_kernel<2, false, false><<<gN, blk, 0, stream>>>(
        nullptr, agg, nfeat, deg, nullptr, nullptr, nullptr,
        nullptr, nullptr, nullptr, wn1, bn1, h1, N, 4 * D, D, D, lg);
    // he1 (raw) = [hn1[src] | efeat | tsenc | hn1[dst]] @ we1 + be1   (overwrites TE-hidden)
    gemm_kernel<3, false, false><<<gE, blk, 0, stream>>>(
        nullptr, nullptr, nullptr, nullptr, h1, efeat, tsenc,
        src, dst, nullptr, we1, be1, e1, E, 4 * D, D, D, lg);

    // ---- layer 2 (relu applied on read of h1/e1) ----
    fill_kernel<<<2048, 256, 0, stream>>>(agg, 0.0f, (size_t)N * 3 * D);
    scatter_kernel<true><<<E, 256, 0, stream>>>(h1, e1, tsenc, src, dst, agg, E, D);
    gemm_kernel<2, false, true><<<gN, blk, 0, stream>>>(
        nullptr, agg, h1, deg, nullptr, nullptr, nullptr,
        nullptr, nullptr, nullptr, wn2, bn2, h2, N, 4 * D, D, D, lg);
    // final: he2 rows [0, n_train) straight into d_out (no relu)
    gemm_kernel<3, false, true><<<gT, blk, 0, stream>>>(
        nullptr, nullptr, nullptr, nullptr, h2, e1, tsenc,
        src, dst, nullptr, we2, be2, out, n_train, 4 * D, D, D, lg);
}